// GcnIdStraight_7919919694203
// MI455X (gfx1250) — compile-verified
//
#include <hip/hip_runtime.h>

// ---------------------------------------------------------------------------
// GCN 2-layer forward for MI455X (gfx1250, wave32).
//   x:[N,64] f32, edge_index:[2,E] i32, w:[E] f32, W1/W2:[64,64], b1/b2:[64]
//   out = relu( S(S(x W1)+b1) W2 + b2 ),  S = norm-weighted scatter over edges
// GEMM uses V_WMMA_F32_16X16X4_F32 (exact fp32 matrix path).
// Edge phases use float4 + native f32 global atomics (L2-resident, 12.8 MB
// node matrices << 192 MB L2).
// ---------------------------------------------------------------------------

typedef __attribute__((ext_vector_type(2))) float v2f;
typedef __attribute__((ext_vector_type(8))) float v8f;

constexpr int FEAT = 64;   // feature width
constexpr int CT   = 4;    // column tiles of 16 (64/16)

__global__ void zero_f32_kernel(float* __restrict__ p, int n) {
  int i = blockIdx.x * blockDim.x + threadIdx.x;
  if (i < n) p[i] = 0.0f;
}

// deg[col[e]] += w[e]
__global__ void deg_kernel(const int* __restrict__ col, const float* __restrict__ w,
                           float* __restrict__ deg, int n_edges) {
  int e = blockIdx.x * blockDim.x + threadIdx.x;
  if (e < n_edges) unsafeAtomicAdd(&deg[col[e]], w[e]);
}

// deg -> deg^{-1/2} in place (0 where deg<=0)
__global__ void rsqrt_kernel(float* __restrict__ deg, int n) {
  int i = blockIdx.x * blockDim.x + threadIdx.x;
  if (i < n) {
    float d = deg[i];
    deg[i] = (d > 0.0f) ? rsqrtf(d) : 0.0f;
  }
}

// norm[e] = dis[row]*w*dis[col]
__global__ void norm_kernel(const int* __restrict__ row, const int* __restrict__ col,
                            const float* __restrict__ w, const float* __restrict__ dis,
                            float* __restrict__ norm, int n_edges) {
  int e = blockIdx.x * blockDim.x + threadIdx.x;
  if (e < n_edges) norm[e] = dis[row[e]] * w[e] * dis[col[e]];
}

// H[N,64] = X[N,64] @ W[64,64], one wave per 16x16 output tile,
// K marched in steps of 4 via V_WMMA_F32_16X16X4_F32.
// A layout (16x4 f32): lanes 0-15 -> K=k0,k0+1 ; lanes 16-31 -> K=k0+2,k0+3.
// B layout (4x16 f32): same K split, N = lane%16.
// C/D layout: vgpr v, lane l -> M = v + 8*(l>=16), N = l%16.
__global__ void gemm64_wmma_kernel(const float* __restrict__ X, const float* __restrict__ W,
                                   float* __restrict__ H, int n_rows) {
  int wave = (blockIdx.x * blockDim.x + threadIdx.x) >> 5;
  int lane = threadIdx.x & 31;
  int row_tile = wave >> 2;          // wave / CT
  int col_tile = wave & (CT - 1);
  int row_base = row_tile * 16;
  if (row_base >= n_rows) return;    // wave-uniform exit: EXEC stays all-1s for WMMA

  int half = lane >> 4;              // 0: lanes 0-15, 1: lanes 16-31
  int m    = lane & 15;
  int cb   = col_tile * 16;

  int arow = row_base + m;
  if (arow > n_rows - 1) arow = n_rows - 1;   // clamp (keeps EXEC full)
  const float* xrow = X + (size_t)arow * FEAT;

  v8f acc = {};
#pragma unroll
  for (int k0 = 0; k0 < FEAT; k0 += 4) {
    int k = k0 + 2 * half;
    v2f a, b;
    a.x = xrow[k];
    a.y = xrow[k + 1];
    b.x = W[(size_t)k * FEAT + cb + m];
    b.y = W[(size_t)(k + 1) * FEAT + cb + m];
    acc = __builtin_amdgcn_wmma_f32_16x16x4_f32(
        /*neg_a=*/false, a, /*neg_b=*/false, b,
        /*c_mod=*/(short)0, acc, /*reuse_a=*/false, /*reuse_b=*/false);
  }
#pragma unroll
  for (int v = 0; v < 8; ++v) {
    int r = row_base + v + 8 * half;
    if (r < n_rows) H[(size_t)r * FEAT + cb + m] = acc[v];
  }
}

// out[n][f] = b[f]  (bias pre-seeded into scatter accumulator)
__global__ void init_bias_kernel(float* __restrict__ out, const float* __restrict__ b, int total) {
  int i = blockIdx.x * blockDim.x + threadIdx.x;
  if (i < total) out[i] = b[i & (FEAT - 1)];
}

// out[col[e]] += H[row[e]] * norm[e]; 16 threads per edge, float4 per thread
__global__ void scatter_kernel(const float* __restrict__ H, const int* __restrict__ row,
                               const int* __restrict__ col, const float* __restrict__ norm,
                               float* __restrict__ out, int n_edges) {
  int t = blockIdx.x * blockDim.x + threadIdx.x;
  int e = t >> 4;
  if (e >= n_edges) return;
  int f  = (t & 15) << 2;
  int r  = row[e];
  int c  = col[e];
  float nv = norm[e];
  const float4* hp = (const float4*)(H + (size_t)r * FEAT + f);
  float4 h = *hp;
  float* op = out + (size_t)c * FEAT + f;
  unsafeAtomicAdd(op + 0, h.x * nv);
  unsafeAtomicAdd(op + 1, h.y * nv);
  unsafeAtomicAdd(op + 2, h.z * nv);
  unsafeAtomicAdd(op + 3, h.w * nv);
}

__global__ void relu_inplace_kernel(float* __restrict__ p, int n) {
  int i = blockIdx.x * blockDim.x + threadIdx.x;
  if (i < n) p[i] = fmaxf(p[i], 0.0f);
}

extern "C" void kernel_launch(void* const* d_in, const int* in_sizes, int n_in,
                              void* d_out, int out_size, void* d_ws, size_t ws_size,
                              hipStream_t stream) {
  const float* x  = (const float*)d_in[0];
  const int*   ei = (const int*)d_in[1];   // [2, E] int32 (JAX x64-off demotes int64)
  const float* w  = (const float*)d_in[2];
  const float* W1 = (const float*)d_in[3];
  const float* b1 = (const float*)d_in[4];
  const float* W2 = (const float*)d_in[5];
  const float* b2 = (const float*)d_in[6];

  const int n_nodes = in_sizes[0] / FEAT;
  const int n_edges = in_sizes[1] / 2;
  const int* row = ei;
  const int* col = ei + n_edges;

  // Workspace carve-up (fp32, 16B-aligned chunks)
  float* ws  = (float*)d_ws;
  size_t off = 0;
  auto carve = [&](size_t nf) { float* p = ws + off; off += (nf + 3) & ~(size_t)3; return p; };
  float* deg  = carve((size_t)n_nodes);             // then rsqrt in place
  float* norm = carve((size_t)n_edges);
  float* h1   = carve((size_t)n_nodes * FEAT);
  float* agg1 = carve((size_t)n_nodes * FEAT);
  float* h2   = carve((size_t)n_nodes * FEAT);
  float* outp = (float*)d_out;

  const int B = 256;
  auto nb = [&](long long n) { return (unsigned)((n + B - 1) / B); };

  const long long nodeF     = (long long)n_nodes * FEAT;
  const long long edgeT     = (long long)n_edges * 16;
  const long long gemmT     = (long long)((n_nodes + 15) / 16) * CT * 32;

  // --- normalization ---
  zero_f32_kernel <<<nb(n_nodes), B, 0, stream>>>(deg, n_nodes);
  deg_kernel      <<<nb(n_edges), B, 0, stream>>>(col, w, deg, n_edges);
  rsqrt_kernel    <<<nb(n_nodes), B, 0, stream>>>(deg, n_nodes);
  norm_kernel     <<<nb(n_edges), B, 0, stream>>>(row, col, w, deg, norm, n_edges);

  // --- layer 1: h1 = x@W1 ; agg1 = b1 + scatter(h1) ---
  gemm64_wmma_kernel<<<nb(gemmT), B, 0, stream>>>(x, W1, h1, n_nodes);
  init_bias_kernel  <<<nb(nodeF), B, 0, stream>>>(agg1, b1, (int)nodeF);
  scatter_kernel    <<<nb(edgeT), B, 0, stream>>>(h1, row, col, norm, agg1, n_edges);

  // --- layer 2: h2 = agg1@W2 ; out = relu(b2 + scatter(h2)) ---
  gemm64_wmma_kernel<<<nb(gemmT), B, 0, stream>>>(agg1, W2, h2, n_nodes);
  init_bias_kernel  <<<nb(nodeF), B, 0, stream>>>(outp, b2, (int)nodeF);
  scatter_kernel    <<<nb(edgeT), B, 0, stream>>>(h2, row, col, norm, outp, n_edges);
  relu_inplace_kernel<<<nb(nodeF), B, 0, stream>>>(outp, (int)nodeF);
}